// GIN_23218593202883
// MI455X (gfx1250) — compile-verified
//
#include <hip/hip_runtime.h>
#include <hip/hip_bf16.h>

// ---------------------------------------------------------------------------
// GIN (2x GINConv + linear head) for MI455X / gfx1250 (wave32, WMMA).
//
// Roofline reasoning (compile-only; reasoning from specs):
//  * x / h / agg are each 100000*128*4B = 51.2 MB -> resident in 192 MB L2.
//  * Scatter phase: per layer 1.6M edges * 512B gather + 512B atomic f32 add,
//    all L2-resident -> L2 bandwidth / atomic-throughput bound. Wave-per-edge,
//    float4 per lane => contiguous 512B gathers, global_atomic_add_f32.
//  * GEMMs ~13 GFLOP fp32 total -> use V_WMMA_F32_16X16X4_F32 (exact fp32
//    semantics matching the reference). Each wave owns a 32x128 C strip
//    (16 WMMA per K-step vs 10 LDS b64 loads -> compute-bound inner loop).
//    __launch_bounds__(128,1) gives the allocator the full VGPR file so the
//    128 accumulator VGPRs never spill (round-3 asm showed scratch spills
//    under the default occupancy-driven budget).
//  * Weights are pre-transposed once to global, then DMA'd into LDS with
//    GLOBAL_LOAD_ASYNC_TO_LDS_B128 (ASYNCcnt path) into a 132-float padded
//    layout that makes the WMMA half-wave K-split bank-conflict free.
//  * BN training stats fused into GEMM epilogue (atomics), BN-apply fused
//    into following elementwise / final GEMV passes.
// ---------------------------------------------------------------------------

#define NNODES 100000
#define NEDGES 1600000
#define CH     128
#define KPAD   132          // 132*4 = 528 B row stride: 16B aligned, and the
                            // half-wave K-split of WMMA frags is conflict-free
#define BN_EPS_C 1e-5f

typedef float v2f __attribute__((ext_vector_type(2)));
typedef float v8f __attribute__((ext_vector_type(8)));
typedef int   v4i_vs __attribute__((vector_size(16)));   // builtin's pointee type

#if defined(__has_builtin)
#if __has_builtin(__builtin_amdgcn_global_load_async_to_lds_b128) && \
    __has_builtin(__builtin_amdgcn_s_wait_asynccnt)
#define HAVE_ASYNC_LDS 1
#endif
#endif

#ifdef HAVE_ASYNC_LDS
// signature (from hipcc diagnostic): (v4i AS1*, v4i AS3*, imm offset, imm cpol)
#define ASYNC_CP16(ldsptr, gp)                                              \
    __builtin_amdgcn_global_load_async_to_lds_b128(                         \
        (__attribute__((address_space(1))) v4i_vs*)(gp),                    \
        (__attribute__((address_space(3))) v4i_vs*)(ldsptr), 0, 0)
#endif

// ---------------------------------------------------------------- utilities
__global__ void zero_f32(float* __restrict__ p, int n) {
    int i = blockIdx.x * blockDim.x + threadIdx.x;
    if (i < n) p[i] = 0.0f;
}

// Wt[n][k] = W[k][n]; 64 KB, L2-resident, done once per GEMM launch.
__global__ void transpose128(const float* __restrict__ W, float* __restrict__ Wt) {
    int i = blockIdx.x * 256 + threadIdx.x;     // 0 .. 16383
    int k = i >> 7, n = i & 127;
    Wt[(size_t)n * CH + k] = W[i];
}

// ------------------------------------------------------------ edge scatter
// agg[dst] += x[src], one wave per edge, float4 per lane (128 ch / 32 lanes).
__global__ void scatter_add(const float* __restrict__ x,
                            const long long* __restrict__ ei,   // [2,E] int64
                            float* __restrict__ agg) {
    int e = blockIdx.x * 8 + (threadIdx.x >> 5);
    if (e >= NEDGES) return;
    int lane = threadIdx.x & 31;
    int src = (int)ei[e];
    int dst = (int)ei[(size_t)NEDGES + e];
    const float4 v = *(const float4*)(x + (size_t)src * CH + lane * 4);
    float* a = agg + (size_t)dst * CH + lane * 4;
    atomicAdd(a + 0, v.x);
    atomicAdd(a + 1, v.y);
    atomicAdd(a + 2, v.z);
    atomicAdd(a + 3, v.w);
}

// ------------------------------------------------------------ WMMA GEMM
// out = relu( in' @ W + b ),  in' = (1+eps)*in + agg   (if agg != nullptr)
// M = nrows, N = K = 128, fp32 via v_wmma_f32_16x16x4_f32.
// Block: 128 thr = 4 waves, 128 rows/block; wave computes a 32x128 C strip
// (two 16-row A tiles -> 16 WMMA per K-step against 10 LDS b64 loads).
// Wt is the PRE-TRANSPOSED weight matrix: Wt[n][k] = W[k][n].
// Optional epilogue: per-channel sum / sum-of-squares atomics (BN stats).
__global__ void __launch_bounds__(128, 1)
gemm_relu_wmma(const float* __restrict__ in,
               const float* __restrict__ agg,   // nullable
               const float* __restrict__ eps,   // nullable
               const float* __restrict__ Wt,    // [128,128] transposed
               const float* __restrict__ bias,  // [128]
               float* __restrict__ out,
               float* __restrict__ stats,       // nullable [256]
               int nrows) {
    extern __shared__ float smem[];
    float* sA = smem;                 // [128][KPAD]  staged A rows
    float* sW = smem + 128 * KPAD;    // [128][KPAD]  sW[n][k] (from Wt)
    float* sB = smem + 2 * 128 * KPAD;// [128]        bias

    const int tid  = threadIdx.x;     // block = 128
    const int lane = tid & 31;
    const int wave = tid >> 5;        // 0..3
    const int m0   = blockIdx.x * 128;

    // ---- stage W: Wt rows -> padded LDS rows (async DMA if available)
#ifdef HAVE_ASYNC_LDS
    for (int idx = tid; idx < CH * 32; idx += 128) {          // 16B chunks
        int n = idx >> 5, c4 = (idx & 31) << 2;
        ASYNC_CP16(&sW[n * KPAD + c4], Wt + (size_t)idx * 4);
    }
#else
    for (int idx = tid; idx < CH * 32; idx += 128) {
        int n = idx >> 5, c4 = (idx & 31) << 2;
        *(float4*)(&sW[n * KPAD + c4]) = *(const float4*)(Wt + (size_t)idx * 4);
    }
#endif

    // ---- stage A block (128 rows x 128 cols)
    if (agg != nullptr) {             // fused (1+eps)*x + agg (needs ALU)
        const float e1 = 1.0f + *eps;
        for (int idx = tid; idx < 128 * 32; idx += 128) {
            int r  = idx >> 5;
            int c4 = (idx & 31) << 2;
            int m  = m0 + r;
            float4 v = make_float4(0.f, 0.f, 0.f, 0.f);
            if (m < nrows) {
                v = *(const float4*)(in + (size_t)m * CH + c4);
                float4 g = *(const float4*)(agg + (size_t)m * CH + c4);
                v.x = e1 * v.x + g.x;  v.y = e1 * v.y + g.y;
                v.z = e1 * v.z + g.z;  v.w = e1 * v.w + g.w;
            }
            *(float4*)(&sA[r * KPAD + c4]) = v;
        }
    } else {                          // plain copy -> async DMA if available
#ifdef HAVE_ASYNC_LDS
        for (int idx = tid; idx < 128 * 32; idx += 128) {
            int r  = idx >> 5;
            int c4 = (idx & 31) << 2;
            if (m0 + r < nrows)       // stale LDS rows belong to skipped waves
                ASYNC_CP16(&sA[r * KPAD + c4],
                           in + (size_t)(m0 + r) * CH + c4);
        }
#else
        for (int idx = tid; idx < 128 * 32; idx += 128) {
            int r  = idx >> 5;
            int c4 = (idx & 31) << 2;
            float4 v = make_float4(0.f, 0.f, 0.f, 0.f);
            if (m0 + r < nrows)
                v = *(const float4*)(in + (size_t)(m0 + r) * CH + c4);
            *(float4*)(&sA[r * KPAD + c4]) = v;
        }
#endif
    }
    if (tid < CH) sB[tid] = bias[tid];
#ifdef HAVE_ASYNC_LDS
    __builtin_amdgcn_s_wait_asynccnt(0);
#endif
    __syncthreads();

    const int mw = m0 + wave * 32;          // this wave's 32-row strip
    if (mw >= nrows) return;                // N % 32 == 0 -> all-or-nothing

    // ISA 32-bit A frag: lane l -> row (l&15), K = kb + ((l>>4)<<1) + {0,1}
    const int rsel = lane & 15;
    const int koff = (lane >> 4) << 1;
    const float* a0p = &sA[(wave * 32 + rsel) * KPAD + koff];
    const float* a1p = a0p + 16 * KPAD;

    v8f acc0[8] = {};
    v8f acc1[8] = {};
    for (int kb = 0; kb < CH; kb += 4) {
        v2f a0 = *(const v2f*)(a0p + kb);
        v2f a1 = *(const v2f*)(a1p + kb);
#pragma unroll
        for (int nt = 0; nt < 8; ++nt) {
            v2f b = *(const v2f*)(&sW[(nt * 16 + rsel) * KPAD + koff + kb]);
            acc0[nt] = __builtin_amdgcn_wmma_f32_16x16x4_f32(
                false, a0, false, b, (short)0, acc0[nt], false, false);
            acc1[nt] = __builtin_amdgcn_wmma_f32_16x16x4_f32(
                false, a1, false, b, (short)0, acc1[nt], false, false);
        }
    }

    // ---- epilogue: bias + relu + store (+ BN stats)
    const int rbase = (lane >> 4) << 3;     // C tile: vgpr r -> row r + 8*(l>>4)
#pragma unroll
    for (int nt = 0; nt < 8; ++nt) {
        const int col = nt * 16 + rsel;
        const float bia = sB[col];
        float s = 0.f, sq = 0.f;
#pragma unroll
        for (int r = 0; r < 8; ++r) {
            float v0 = fmaxf(acc0[nt][r] + bia, 0.0f);
            float v1 = fmaxf(acc1[nt][r] + bia, 0.0f);
            out[(size_t)(mw +      rbase + r) * CH + col] = v0;
            out[(size_t)(mw + 16 + rbase + r) * CH + col] = v1;
            s += v0 + v1; sq += v0 * v0 + v1 * v1;
        }
        if (stats != nullptr) {
            s  += __shfl_down(s, 16);
            sq += __shfl_down(sq, 16);
            if (lane < 16) {
                atomicAdd(&stats[col], s);
                atomicAdd(&stats[CH + col], sq);
            }
        }
    }
}

// ------------------------------------------------------------ BN finalize
// params[c] = gamma*rsqrt(var+eps), params[128+c] = beta - mu*scale
__global__ void bn_finalize(const float* __restrict__ stats,
                            const float* __restrict__ gamma,
                            const float* __restrict__ beta,
                            float* __restrict__ params) {
    int c = threadIdx.x;
    if (c >= CH) return;
    const float invN = 1.0f / (float)NNODES;
    float mu  = stats[c] * invN;
    float var = stats[CH + c] * invN - mu * mu;
    float scale = gamma[c] * rsqrtf(var + BN_EPS_C);
    params[c]      = scale;
    params[CH + c] = beta[c] - mu * scale;
}

// ------------------------------------------------------------ BN apply+relu
__global__ void bn_apply_relu(const float* __restrict__ in,
                              const float* __restrict__ params,
                              float* __restrict__ out) {
    int t = blockIdx.x * blockDim.x + threadIdx.x;
    size_t base = (size_t)t * 4;
    if (base >= (size_t)NNODES * CH) return;
    int c0 = (int)(base & (CH - 1));
    float4 v = *(const float4*)(in + base);
    v.x = fmaxf(v.x * params[c0 + 0] + params[CH + c0 + 0], 0.f);
    v.y = fmaxf(v.y * params[c0 + 1] + params[CH + c0 + 1], 0.f);
    v.z = fmaxf(v.z * params[c0 + 2] + params[CH + c0 + 2], 0.f);
    v.w = fmaxf(v.w * params[c0 + 3] + params[CH + c0 + 3], 0.f);
    *(float4*)(out + base) = v;
}

// ------------------------------------------------- fused BN+relu+out GEMV
// out[i] = sum_c relu(h[i,c]*scale+shift) * Wout[c] + bout ; wave per row.
__global__ void bn_gemv_out(const float* __restrict__ h,
                            const float* __restrict__ params,
                            const float* __restrict__ wout,
                            const float* __restrict__ bout,
                            float* __restrict__ out) {
    int row = blockIdx.x * 8 + (threadIdx.x >> 5);
    if (row >= NNODES) return;
    int lane = threadIdx.x & 31;
    float s = 0.f;
#pragma unroll
    for (int j = 0; j < 4; ++j) {
        int c = lane + j * 32;
        float v = h[(size_t)row * CH + c] * params[c] + params[CH + c];
        s += fmaxf(v, 0.0f) * wout[c];
    }
#pragma unroll
    for (int off = 16; off > 0; off >>= 1) s += __shfl_down(s, off);
    if (lane == 0) out[row] = s + bout[0];
}

// ---------------------------------------------------------------------------
extern "C" void kernel_launch(void* const* d_in, const int* in_sizes, int n_in,
                              void* d_out, int out_size, void* d_ws, size_t ws_size,
                              hipStream_t stream) {
    const float*     x    = (const float*)d_in[0];
    const long long* ei   = (const long long*)d_in[1];
    const float*     eps1 = (const float*)d_in[2];
    const float*     W1a  = (const float*)d_in[3];
    const float*     b1a  = (const float*)d_in[4];
    const float*     W1b  = (const float*)d_in[5];
    const float*     b1b  = (const float*)d_in[6];
    const float*     g1   = (const float*)d_in[7];
    const float*     be1  = (const float*)d_in[8];
    const float*     eps2 = (const float*)d_in[9];
    const float*     W2a  = (const float*)d_in[10];
    const float*     b2a  = (const float*)d_in[11];
    const float*     W2b  = (const float*)d_in[12];
    const float*     b2b  = (const float*)d_in[13];
    const float*     g2   = (const float*)d_in[14];
    const float*     be2  = (const float*)d_in[15];
    const float*     Wout = (const float*)d_in[16];
    const float*     bout = (const float*)d_in[17];
    float*           outp = (float*)d_out;

    const size_t NC = (size_t)NNODES * CH;           // 12.8M floats
    float* A      = (float*)d_ws;                    // N x 128
    float* B      = A + NC;                          // N x 128
    float* stats0 = B + NC;                          // 256
    float* stats1 = stats0 + 256;                    // 256
    float* par0   = stats1 + 256;                    // 256
    float* par1   = par0 + 256;                      // 256
    float* Wt     = par1 + 256;                      // 128x128 transposed W

    const int    zgridNC  = (int)((NC + 255) / 256);
    const int    egrid    = (NEDGES + 7) / 8;
    const int    ggrid    = (NNODES + 127) / 128;
    const int    agrid    = (int)((NC / 4 + 255) / 256);
    const int    rgrid    = (NNODES + 7) / 8;
    const size_t smemB    = (size_t)(2 * 128 * KPAD + 128) * sizeof(float);

    // ---------------- layer 1 ----------------
    zero_f32<<<zgridNC, 256, 0, stream>>>(A, (int)NC);
    zero_f32<<<1, 256, 0, stream>>>(stats0, 256);
    scatter_add<<<egrid, 256, 0, stream>>>(x, ei, A);
    transpose128<<<64, 256, 0, stream>>>(W1a, Wt);
    gemm_relu_wmma<<<ggrid, 128, smemB, stream>>>(x, A, eps1, Wt, b1a, B,
                                                  nullptr, NNODES);
    transpose128<<<64, 256, 0, stream>>>(W1b, Wt);
    gemm_relu_wmma<<<ggrid, 128, smemB, stream>>>(B, nullptr, nullptr, Wt, b1b,
                                                  A, stats0, NNODES);
    bn_finalize<<<1, 128, 0, stream>>>(stats0, g1, be1, par0);
    bn_apply_relu<<<agrid, 256, 0, stream>>>(A, par0, B);   // B = h1

    // ---------------- layer 2 ----------------
    zero_f32<<<zgridNC, 256, 0, stream>>>(A, (int)NC);
    zero_f32<<<1, 256, 0, stream>>>(stats1, 256);
    scatter_add<<<egrid, 256, 0, stream>>>(B, ei, A);
    transpose128<<<64, 256, 0, stream>>>(W2a, Wt);
    gemm_relu_wmma<<<ggrid, 128, smemB, stream>>>(B, A, eps2, Wt, b2a, B,
                                                  nullptr, NNODES);   // in-place safe: rows staged before stores
    transpose128<<<64, 256, 0, stream>>>(W2b, Wt);
    gemm_relu_wmma<<<ggrid, 128, smemB, stream>>>(B, nullptr, nullptr, Wt, b2b,
                                                  A, stats1, NNODES);
    bn_finalize<<<1, 128, 0, stream>>>(stats1, g2, be2, par1);

    // ---------------- head ----------------
    bn_gemv_out<<<rgrid, 256, 0, stream>>>(A, par1, Wout, bout, outp);
}